// DenseGrid_63806034149465
// MI455X (gfx1250) — compile-verified
//
#include <hip/hip_runtime.h>
#include <hip/hip_bf16.h>
#include <stdint.h>

// ---------------------------------------------------------------------------
// DenseGrid 1-D multiresolution lerp encoding (N=1M, L=16, F=4).
// Store-bandwidth bound: ~260 MB traffic -> ~11 us at 23.3 TB/s. No matmul, so
// no WMMA role; CDNA5 paths used: async global->LDS table staging (ASYNCcnt +
// s_wait_asynccnt), NT streaming b128 stores, wave32-coalesced output layout.
// This revision minimizes instructions per output byte (issue-rate headroom).
// ---------------------------------------------------------------------------

typedef float v4f __attribute__((ext_vector_type(4)));
typedef int   v4i __attribute__((vector_size(4 * sizeof(int))));

typedef __attribute__((address_space(1))) v4i* gptr_v4i;  // global int4*
typedef __attribute__((address_space(3))) v4i* lptr_v4i;  // LDS int4*

#define LEVELS      16
#define MAX_ROWS    320                  // sum of resolutions [5,7,...,35]
#define TPB         256                  // 8 wave32 per block
#define PTS_PER_IT  (TPB / LEVELS)       // 16 points per iteration
#define ITERS       32
#define PPB         (PTS_PER_IT * ITERS) // 512 points per block

#if __has_builtin(__builtin_amdgcn_global_load_async_to_lds_b128)
#define HAVE_ASYNC_BUILTIN 1
#else
#define HAVE_ASYNC_BUILTIN 0
#endif
#if __has_builtin(__builtin_amdgcn_s_wait_asynccnt)
#define HAVE_WAITASYNC_BUILTIN 1
#else
#define HAVE_WAITASYNC_BUILTIN 0
#endif

__device__ __forceinline__ void async_copy16_to_lds(const float* gsrc, void* lds_dst) {
#if HAVE_ASYNC_BUILTIN
  __builtin_amdgcn_global_load_async_to_lds_b128(
      (gptr_v4i)(unsigned long long)(uintptr_t)gsrc,
      (lptr_v4i)(unsigned)(uintptr_t)lds_dst,
      0, 0);
#else
  unsigned lds_off = (unsigned)(uintptr_t)lds_dst;
  unsigned long long ga = (unsigned long long)(uintptr_t)gsrc;
  asm volatile("global_load_async_to_lds_b128 %0, %1, off"
               :: "v"(lds_off), "v"(ga) : "memory");
#endif
}

__device__ __forceinline__ void wait_async0() {
#if HAVE_WAITASYNC_BUILTIN
  __builtin_amdgcn_s_wait_asynccnt(0);
#else
  asm volatile("s_wait_asynccnt 0x0" ::: "memory");
#endif
}

__global__ __launch_bounds__(TPB) void densegrid_lerp_kernel(
    const float* __restrict__ x,            // [N]
    const float* __restrict__ storage,      // [total_rows, 4]
    const int*   __restrict__ resolutions,  // [LEVELS]
    float*       __restrict__ out,          // [N, LEVELS, 4]
    int n_points, int total_rows)
{
  __shared__ v4f   s_tab[MAX_ROWS];   // 5120 B staged table
  __shared__ float s_resm1[LEVELS];
  __shared__ int   s_off[LEVELS];

  const int tid = threadIdx.x;

  // ---- stage table into LDS with CDNA5 async b128 copies ----
  const int rows = total_rows < MAX_ROWS ? total_rows : MAX_ROWS;
  for (int c = tid; c < rows; c += TPB)
    async_copy16_to_lds(storage + 4 * c, (void*)&s_tab[c]);

  // ---- per-level (res-1) and row offsets (once per block, amortized) ----
  if (tid < LEVELS) {
    int off = 0;
    for (int j = 0; j < tid; ++j) off += resolutions[j];
    s_off[tid]   = off;
    s_resm1[tid] = (float)(resolutions[tid] - 1);
  }

  wait_async0();       // this wave's async copies landed in LDS
  __syncthreads();     // all waves' copies + offset setup visible

  // lane -> (point-slot, level): flat float4 output index = base*16 + tid,
  // so every wave32 b128 store covers one contiguous 512 B span.
  const int   l    = tid & (LEVELS - 1);
  const int   sub  = tid >> 4;              // 0..15
  const float rm1  = s_resm1[l];
  const v4f* __restrict__ tabp = s_tab + s_off[l];   // loop-invariant LDS base

  const int base = blockIdx.x * PPB;
  const float* __restrict__ xin  = x + base + sub;                 // + it*16
  v4f* __restrict__ outp = (v4f*)out + (size_t)base * LEVELS + tid; // + it*256

  if (base + PPB <= n_points) {
    // fast path: whole block in range, no per-iteration predication
#pragma unroll
    for (int it = 0; it < ITERS; ++it) {
      const float scaled = xin[it * PTS_PER_IT] * rm1;
      const int   i0     = (int)scaled;             // trunc == floor (scaled>=0)
      const float t      = 1.0f - (scaled - (float)i0);
      const v4f   g0     = tabp[i0];                // ds_load_b128
      const v4f   g1     = tabp[i0 + 1];            // ds_load_b128 offset:+16
      const v4f   r      = g0 * t + g1 * (1.0f - t);
      __builtin_nontemporal_store(r, outp + it * TPB);
    }
  } else {
#pragma unroll 4
    for (int it = 0; it < ITERS; ++it) {
      const int n = base + it * PTS_PER_IT + sub;
      if (n < n_points) {
        const float scaled = xin[it * PTS_PER_IT] * rm1;
        const int   i0     = (int)scaled;
        const float t      = 1.0f - (scaled - (float)i0);
        const v4f   g0     = tabp[i0];
        const v4f   g1     = tabp[i0 + 1];
        const v4f   r      = g0 * t + g1 * (1.0f - t);
        __builtin_nontemporal_store(r, outp + it * TPB);
      }
    }
  }
}

extern "C" void kernel_launch(void* const* d_in, const int* in_sizes, int n_in,
                              void* d_out, int out_size, void* d_ws, size_t ws_size,
                              hipStream_t stream) {
  const float* x       = (const float*)d_in[0];
  const float* storage = (const float*)d_in[1];
  const int*   res     = (const int*)d_in[2];
  float*       out     = (float*)d_out;
  const int n          = in_sizes[0];        // x is [N,1] -> N floats
  const int total_rows = in_sizes[1] / 4;    // storage is [rows, 4]

  const int blocks = (n + PPB - 1) / PPB;
  densegrid_lerp_kernel<<<blocks, TPB, 0, stream>>>(x, storage, res, out, n, total_rows);
}